// NaiveGATLayer_59081570124187
// MI455X (gfx1250) — compile-verified
//
#include <hip/hip_runtime.h>
#include <hip/hip_bf16.h>
#include <math.h>

#define N_NODES 50000
#define N_EDGES 1600000
#define IN_C    128
#define OUT_C   128
#define HEADS   8
#define HFD     16
#define NEG_SLOPE 0.2f

typedef __attribute__((ext_vector_type(2))) float v2f;
typedef __attribute__((ext_vector_type(8))) float v8f;

// ---------------------------------------------------------------------------
// Kernel 1: emb[N,128] = X[N,128] @ W[128,128]^T  via fp32 WMMA 16x16x4.
// One wave computes one 16x16 output tile; 8 waves/block cover the 8 column
// tiles of a 16-row stripe. grid = N/16 = 3125 blocks.
// Fragment layout (ISA 7.12.2, 32-bit A 16x4 / B 4x16):
//   lane L: half = L>=16, m/n = L&15
//   A frag = { X[mBase+m][k0+2*half], X[mBase+m][k0+2*half+1] }
//   B frag = { B[k0+2*half][oBase+n], B[k0+2*half+1][oBase+n] }
//          = { W[oBase+n][k0+2*half], W[oBase+n][k0+2*half+1] }   (B = W^T)
//   D elem r -> row mBase + r + 8*half, col oBase + n
// ---------------------------------------------------------------------------
__global__ __launch_bounds__(256) void gat_gemm_wmma(
    const float* __restrict__ X,     // [N, IN_C]
    const float* __restrict__ W,     // [OUT_C, IN_C] row-major
    float* __restrict__ emb)         // [N, OUT_C]
{
    const int wave = threadIdx.x >> 5;   // 0..7 -> output column tile
    const int lane = threadIdx.x & 31;
    const int half = lane >> 4;          // 0 | 1
    const int l16  = lane & 15;
    const int mBase = blockIdx.x * 16;
    const int oBase = wave * 16;

    const float* aPtr = X + (size_t)(mBase + l16) * IN_C + 2 * half;
    const float* bPtr = W + (size_t)(oBase + l16) * IN_C + 2 * half;

    v8f acc = {};
#pragma unroll
    for (int k0 = 0; k0 < IN_C; k0 += 4) {
        v2f a = *(const v2f*)(aPtr + k0);
        v2f b = *(const v2f*)(bPtr + k0);
        acc = __builtin_amdgcn_wmma_f32_16x16x4_f32(
            /*neg_a=*/false, a, /*neg_b=*/false, b,
            /*c_mod=*/(short)0, acc, /*reuse_a=*/false, /*reuse_b=*/false);
    }

    float* o = emb + (size_t)(mBase + 8 * half) * OUT_C + oBase + l16;
#pragma unroll
    for (int r = 0; r < 8; ++r)
        o[(size_t)r * OUT_C] = acc[r];
}

// ---------------------------------------------------------------------------
// Kernel 2: per-node logit halves  left/right[n,h] = sum_c emb[n,h*16+c]*a[c,h]
// One thread per (n,h): 64B contiguous emb read, a_left/a_right are 512B,
// fully cache-resident.
// ---------------------------------------------------------------------------
__global__ void gat_logits(const float* __restrict__ emb,
                           const float* __restrict__ aL,   // [HFD, HEADS]
                           const float* __restrict__ aR,   // [HFD, HEADS]
                           float* __restrict__ left,       // [N, HEADS]
                           float* __restrict__ right)      // [N, HEADS]
{
    int idx = blockIdx.x * blockDim.x + threadIdx.x;       // n*HEADS + h
    if (idx >= N_NODES * HEADS) return;
    int h = idx & (HEADS - 1);
    int n = idx >> 3;
    const float* ev = emb + (size_t)n * OUT_C + h * HFD;
    float sl = 0.f, sr = 0.f;
#pragma unroll
    for (int c = 0; c < HFD; ++c) {
        float e = ev[c];
        sl += e * aL[c * HEADS + h];
        sr += e * aR[c * HEADS + h];
    }
    left[idx]  = sl;
    right[idx] = sr;
}

// ---------------------------------------------------------------------------
// Kernel 3: init out = bias (so atomic aggregation lands on final result)
// and denom = 0.
// ---------------------------------------------------------------------------
__global__ void gat_init(float* __restrict__ out,
                         float* __restrict__ denom,
                         const float* __restrict__ bias)
{
    int idx = blockIdx.x * blockDim.x + threadIdx.x;
    if (idx < N_NODES * OUT_C)  out[idx]   = bias[idx & (OUT_C - 1)];
    if (idx < N_NODES * HEADS)  denom[idx] = 0.f;
}

// ---------------------------------------------------------------------------
// Kernel 4: softmax denominator.  One thread per (edge, head).
// left/right/denom are L2-resident (1.6 MB each); edge list streams from HBM.
// ---------------------------------------------------------------------------
__global__ void gat_denom(const int* __restrict__ src,
                          const int* __restrict__ dst,
                          const float* __restrict__ left,
                          const float* __restrict__ right,
                          float* __restrict__ denom)
{
    unsigned idx = blockIdx.x * blockDim.x + threadIdx.x;  // e*HEADS + h
    if (idx >= (unsigned)N_EDGES * HEADS) return;
    unsigned e = idx >> 3;
    int h = idx & (HEADS - 1);
    int s = src[e];
    int d = dst[e];
    float x = left[s * HEADS + h] + right[d * HEADS + h];
    x = (x >= 0.f) ? x : NEG_SLOPE * x;
    unsafeAtomicAdd(&denom[d * HEADS + h], expf(x));
}

// ---------------------------------------------------------------------------
// Kernel 5: weighted aggregation.  One thread per (edge, channel):
// 128 consecutive threads share an edge (broadcast src/dst/logit loads,
// coalesced 512B emb row gather), then scatter-add into out (L2 atomics).
// Recomputes exp(lrelu(.)) instead of storing w[E,8] (saves 51 MB of ws).
// ---------------------------------------------------------------------------
__global__ void gat_aggregate(const int* __restrict__ src,
                              const int* __restrict__ dst,
                              const float* __restrict__ left,
                              const float* __restrict__ right,
                              const float* __restrict__ denom,
                              const float* __restrict__ emb,
                              float* __restrict__ out)
{
    unsigned idx = blockIdx.x * blockDim.x + threadIdx.x;  // e*OUT_C + c
    if (idx >= (unsigned)N_EDGES * OUT_C) return;
    unsigned e = idx >> 7;
    int c = idx & (OUT_C - 1);
    int h = c >> 4;
    int s = src[e];
    int d = dst[e];
    float x = left[s * HEADS + h] + right[d * HEADS + h];
    x = (x >= 0.f) ? x : NEG_SLOPE * x;
    float attn = expf(x) / denom[d * HEADS + h];
    float val  = emb[(size_t)s * OUT_C + c] * attn;
    unsafeAtomicAdd(&out[(size_t)d * OUT_C + c], val);
}

// ---------------------------------------------------------------------------
// Launch
// ---------------------------------------------------------------------------
extern "C" void kernel_launch(void* const* d_in, const int* in_sizes, int n_in,
                              void* d_out, int out_size, void* d_ws, size_t ws_size,
                              hipStream_t stream)
{
    const float* X    = (const float*)d_in[0];   // node_feats [N, IN_C]
    const int*   ei   = (const int*)  d_in[1];   // edge_index [2, E]
    const float* W    = (const float*)d_in[2];   // [OUT_C, IN_C]
    const float* aL   = (const float*)d_in[3];   // [HFD, HEADS]
    const float* aR   = (const float*)d_in[4];   // [HFD, HEADS]
    const float* bias = (const float*)d_in[5];   // [OUT_C]
    float* out = (float*)d_out;                  // [N, OUT_C]

    // workspace layout (floats): emb | left | right | denom  (~30.4 MB)
    float* emb   = (float*)d_ws;
    float* left  = emb   + (size_t)N_NODES * OUT_C;
    float* right = left  + (size_t)N_NODES * HEADS;
    float* denom = right + (size_t)N_NODES * HEADS;

    const int* src = ei;
    const int* dst = ei + N_EDGES;

    gat_gemm_wmma<<<N_NODES / 16, 256, 0, stream>>>(X, W, emb);

    gat_logits<<<(N_NODES * HEADS + 255) / 256, 256, 0, stream>>>(
        emb, aL, aR, left, right);

    gat_init<<<(N_NODES * OUT_C + 255) / 256, 256, 0, stream>>>(
        out, denom, bias);

    gat_denom<<<(int)(((size_t)N_EDGES * HEADS + 255) / 256), 256, 0, stream>>>(
        src, dst, left, right, denom);

    gat_aggregate<<<(int)(((size_t)N_EDGES * OUT_C + 255) / 256), 256, 0, stream>>>(
        src, dst, left, right, denom, emb, out);
}